// GCNModelVAE_76630806495674
// MI455X (gfx1250) — compile-verified
//
#include <hip/hip_runtime.h>
#include <hip/hip_bf16.h>

// ---------------------------------------------------------------------------
// GCN-VAE forward for MI455X (gfx1250, wave32, WMMA).
//   xw1 = x @ w1                 fp32 WMMA, B panel staged in LDS via
//                                global_load_async_to_lds (ASYNCcnt path),
//                                K-pair interleaved + bank-padded layout so
//                                each B fragment is one ds_load_b64
//   h1  = spmm(adj, xw1)         wave-per-edge, f32 global atomics
//   hm/hl = relu(h1) @ w_{mean,logstd}   (relu fused into A-load)
//   z_mean/z_log_std = spmm(adj, hm/hl)
//   z   = z_mean + eps*exp(z_log_std)
//   recon = z @ z^T              store-bound: fp32 WMMA + nontemporal stores
// ---------------------------------------------------------------------------

#define NN 8192
#define FF 512
#define H1D 256
#define H2D 128
#define EE 262144

typedef float v2f __attribute__((ext_vector_type(2)));
typedef float v8f __attribute__((ext_vector_type(8)));

#if __has_builtin(__builtin_amdgcn_global_load_async_to_lds_b32) && \
    __has_builtin(__builtin_amdgcn_s_wait_asynccnt)
#define USE_ASYNC_LDS 1
typedef __attribute__((address_space(1))) int g_i32;
typedef __attribute__((address_space(3))) int l_i32;
#endif

// ---------------------------------------------------------------------------
// Generic 16(M) x 64(N) fp32 WMMA tile, B read from global (row-major or
// transposed/contiguous). Used by gemm2 + decoder.
// A frag (ISA 7.12.2, 32-bit A 16x4): lane<16 -> m=lane,k={0,1}; lane>=16 ->
// m=lane-16,k={2,3} => contiguous float2 per lane. B frag mirrored.
// C/D: VGPR r: lanes0-15 -> M=r, lanes16-31 -> M=r+8, N=lane&15.
// ---------------------------------------------------------------------------
template <bool RELU_A, bool B_TRANS, bool NT_STORE>
__device__ __forceinline__ void wmma_tile_16x64(
    const float* __restrict__ A, int lda,
    const float* __restrict__ B, int ldb,
    float* __restrict__ C, long long ldc, int K)
{
  const int lane = threadIdx.x & 31;
  const int r16  = lane & 15;
  const int kh   = lane >> 4;

  v8f acc0 = {}, acc1 = {}, acc2 = {}, acc3 = {};

  const float* ap = A + (size_t)r16 * lda + kh * 2;
  const float* bp = B_TRANS ? (B + (size_t)r16 * ldb + kh * 2)
                            : (B + (size_t)(kh * 2) * ldb + r16);

  for (int k0 = 0; k0 < K; k0 += 4) {
    v2f a = *(const v2f*)ap;
    if (RELU_A) { a.x = fmaxf(a.x, 0.f); a.y = fmaxf(a.y, 0.f); }

    v2f b0, b1, b2, b3;
    if (B_TRANS) {
      b0 = *(const v2f*)(bp);
      b1 = *(const v2f*)(bp + 16 * (size_t)ldb);
      b2 = *(const v2f*)(bp + 32 * (size_t)ldb);
      b3 = *(const v2f*)(bp + 48 * (size_t)ldb);
    } else {
      b0.x = bp[0];  b0.y = bp[ldb + 0];
      b1.x = bp[16]; b1.y = bp[ldb + 16];
      b2.x = bp[32]; b2.y = bp[ldb + 32];
      b3.x = bp[48]; b3.y = bp[ldb + 48];
    }

    acc0 = __builtin_amdgcn_wmma_f32_16x16x4_f32(false, a, false, b0, (short)0, acc0, false, false);
    acc1 = __builtin_amdgcn_wmma_f32_16x16x4_f32(false, a, false, b1, (short)0, acc1, false, false);
    acc2 = __builtin_amdgcn_wmma_f32_16x16x4_f32(false, a, false, b2, (short)0, acc2, false, false);
    acc3 = __builtin_amdgcn_wmma_f32_16x16x4_f32(false, a, false, b3, (short)0, acc3, false, false);

    ap += 4;
    bp += B_TRANS ? 4 : (size_t)4 * ldb;
  }

#pragma unroll
  for (int r = 0; r < 8; ++r) {
    float* cp = C + (long long)(r + kh * 8) * ldc + r16;
    if (NT_STORE) {
      __builtin_nontemporal_store(acc0[r], cp +  0);
      __builtin_nontemporal_store(acc1[r], cp + 16);
      __builtin_nontemporal_store(acc2[r], cp + 32);
      __builtin_nontemporal_store(acc3[r], cp + 48);
    } else {
      cp[ 0] = acc0[r];
      cp[16] = acc1[r];
      cp[32] = acc2[r];
      cp[48] = acc3[r];
    }
  }
}

// ---------------------------------------------------------------------------
// GEMM1: xw1[8192,256] = x[8192,512] @ w1[512,256].
// Block = 256 thr (8 waves) -> 128(M) x 64(N) tile. w1 K-chunks (128x64)
// staged in LDS, K-pair interleaved: Bs[kp*160 + n*2 + (k&1)] so each WMMA
// B-fragment {B[k][n],B[k+1][n]} is one contiguous ds_load_b64. Row stride
// 160 floats == 32 mod 64 banks => half-waves (kp vs kp+1) hit complementary
// bank sets: conflict-free. 64 kp-rows * 640 B = 40 KB (one DS base reg).
// ---------------------------------------------------------------------------
__global__ __launch_bounds__(256) void gemm1_kernel(
    const float* __restrict__ x, const float* __restrict__ w1,
    float* __restrict__ xw1)
{
  __shared__ float Bs[64 * 160];                 // 40 KB K-chunk panel

  const int tid  = threadIdx.x;
  const int n0   = blockIdx.y * 64;
  const int m0   = blockIdx.x * 128 + (tid >> 5) * 16;
  const int lane = tid & 31;
  const int r16  = lane & 15;
  const int kh   = lane >> 4;

  v8f acc0 = {}, acc1 = {}, acc2 = {}, acc3 = {};

  const float* ap = x + (size_t)(m0 + r16) * FF + kh * 2;

  for (int kc = 0; kc < FF; kc += 128) {
    __syncthreads();                             // panel consumed by all waves
    // Stage 128x64 elements; thread t -> i = t + 256*j : k = i>>6 (row),
    // n = i&63 (col; n-fast => coalesced global reads).
#pragma unroll
    for (int j = 0; j < 32; ++j) {
      const int i = tid + 256 * j;
      const int k = i >> 6;
      const int n = i & 63;
      const float* srcp = w1 + (size_t)(kc + k) * H1D + n0 + n;
      float*       dstp = &Bs[(k >> 1) * 160 + n * 2 + (k & 1)];
#ifdef USE_ASYNC_LDS
      __builtin_amdgcn_global_load_async_to_lds_b32((g_i32*)srcp, (l_i32*)dstp, 0, 0);
#else
      *dstp = *srcp;
#endif
    }
#ifdef USE_ASYNC_LDS
    __builtin_amdgcn_s_wait_asynccnt(0);
#endif
    __syncthreads();

    for (int k0 = 0; k0 < 128; k0 += 4) {
      v2f a = *(const v2f*)(ap + kc + k0);
      const int kp = (k0 >> 1) + kh;             // 2*kp == k0 + kh*2 == ka
      const float* bq = &Bs[kp * 160 + r16 * 2];
      const v2f b0 = *(const v2f*)(bq);
      const v2f b1 = *(const v2f*)(bq + 32);
      const v2f b2 = *(const v2f*)(bq + 64);
      const v2f b3 = *(const v2f*)(bq + 96);

      acc0 = __builtin_amdgcn_wmma_f32_16x16x4_f32(false, a, false, b0, (short)0, acc0, false, false);
      acc1 = __builtin_amdgcn_wmma_f32_16x16x4_f32(false, a, false, b1, (short)0, acc1, false, false);
      acc2 = __builtin_amdgcn_wmma_f32_16x16x4_f32(false, a, false, b2, (short)0, acc2, false, false);
      acc3 = __builtin_amdgcn_wmma_f32_16x16x4_f32(false, a, false, b3, (short)0, acc3, false, false);
    }
  }

  float* C = xw1 + (size_t)m0 * H1D + n0;
#pragma unroll
  for (int r = 0; r < 8; ++r) {
    float* cp = C + (size_t)(r + kh * 8) * H1D + r16;
    cp[ 0] = acc0[r];
    cp[16] = acc1[r];
    cp[32] = acc2[r];
    cp[48] = acc3[r];
  }
}

// hm = relu(h1) @ w_mean, hl = relu(h1) @ w_logstd.  4 waves/block; sel is
// wave-uniform so EXEC stays all-ones around WMMA.
__global__ __launch_bounds__(128) void gemm2_kernel(
    const float* __restrict__ h1, const float* __restrict__ wm,
    const float* __restrict__ wl, float* __restrict__ hm,
    float* __restrict__ hl)
{
  const int m0  = blockIdx.x * 16;
  const int w   = threadIdx.x >> 5;
  const int sel = w >> 1;
  const int n0  = (w & 1) * 64;
  const float* B = sel ? wl : wm;
  float*       C = sel ? hl : hm;
  wmma_tile_16x64<true, false, false>(
      h1 + (size_t)m0 * H1D, H1D,
      B + n0, H2D,
      C + (size_t)m0 * H2D + n0, H2D, H1D);
}

// recon = z @ z^T : 268 MB output stream -> NT stores; z is L2-resident.
__global__ __launch_bounds__(256) void decoder_kernel(
    const float* __restrict__ z, float* __restrict__ recon)
{
  const int m0 = blockIdx.x * 16;
  const long long j0 = ((long long)blockIdx.y * 8 + (threadIdx.x >> 5)) * 64;
  wmma_tile_16x64<false, true, true>(
      z + (size_t)m0 * H2D, H2D,
      z + (size_t)j0 * H2D, H2D,
      recon + (long long)m0 * NN + j0, NN, H2D);
}

// ---------------------------------------------------------------------------
// SpMM: out[dst] += val * h[src].  One wave per edge: indices loaded once per
// edge; lanes stride over float4 feature chunks -> coalesced gathers/atomics.
// ---------------------------------------------------------------------------
template <int FDIM>
__global__ __launch_bounds__(256) void spmm_kernel(
    const float* __restrict__ vals, const int* __restrict__ src,
    const int* __restrict__ dst, const float* __restrict__ h,
    float* __restrict__ out)
{
  const int lane = threadIdx.x & 31;
  const int e    = (blockIdx.x * 256 + threadIdx.x) >> 5;  // wave index = edge
  const float v  = vals[e];
  const int   s  = src[e];
  const int   d  = dst[e];
  const float4* hp = (const float4*)(h + (size_t)s * FDIM);
  float*        op = out + (size_t)d * FDIM;
#pragma unroll
  for (int c = lane; c < FDIM / 4; c += 32) {
    const float4 hv = hp[c];
    atomicAdd(op + c * 4 + 0, v * hv.x);
    atomicAdd(op + c * 4 + 1, v * hv.y);
    atomicAdd(op + c * 4 + 2, v * hv.z);
    atomicAdd(op + c * 4 + 3, v * hv.w);
  }
}

__global__ void zero4_kernel(float4* __restrict__ p, int n4)
{
  const int i = blockIdx.x * blockDim.x + threadIdx.x;
  if (i < n4) p[i] = make_float4(0.f, 0.f, 0.f, 0.f);
}

// z = z_mean + eps * exp(z_log_std)
__global__ void reparam_kernel(
    const float* __restrict__ zm, const float* __restrict__ zls,
    const float* __restrict__ eps, float* __restrict__ z, int n)
{
  const int i = blockIdx.x * blockDim.x + threadIdx.x;
  if (i < n) z[i] = zm[i] + eps[i] * expf(zls[i]);
}

extern "C" void kernel_launch(void* const* d_in, const int* in_sizes, int n_in,
                              void* d_out, int out_size, void* d_ws, size_t ws_size,
                              hipStream_t stream)
{
  const float* x    = (const float*)d_in[0];
  const float* vals = (const float*)d_in[1];
  const int*   src  = (const int*)d_in[2];
  const int*   dst  = (const int*)d_in[3];
  const float* w1   = (const float*)d_in[4];
  const float* wm   = (const float*)d_in[5];
  const float* wl   = (const float*)d_in[6];
  const float* eps  = (const float*)d_in[7];

  float* out    = (float*)d_out;
  float* recon  = out;                                    // [N*N]
  float* z_mean = out + (size_t)NN * NN;                  // [N,H2]
  float* z_lstd = z_mean + (size_t)NN * H2D;              // [N,H2]

  float* xw1 = (float*)d_ws;                              // [N,H1]  8 MB
  float* h1  = xw1 + (size_t)NN * H1D;                    // [N,H1]  8 MB
  float* hm  = h1  + (size_t)NN * H1D;                    // [N,H2]  4 MB
  float* hl  = hm  + (size_t)NN * H2D;                    // [N,H2]  4 MB
  float* z   = hl  + (size_t)NN * H2D;                    // [N,H2]  4 MB

  // 1) xw1 = x @ w1  (async-to-LDS staged B panel)
  gemm1_kernel<<<dim3(NN / 128, H1D / 64), 256, 0, stream>>>(x, w1, xw1);

  // 2) h1 = spmm(adj, xw1)
  zero4_kernel<<<(NN * H1D / 4 + 255) / 256, 256, 0, stream>>>((float4*)h1, NN * H1D / 4);
  spmm_kernel<H1D><<<EE * 32 / 256, 256, 0, stream>>>(vals, src, dst, xw1, h1);

  // 3) hm = relu(h1) @ w_mean, hl = relu(h1) @ w_logstd
  gemm2_kernel<<<NN / 16, 128, 0, stream>>>(h1, wm, wl, hm, hl);

  // 4) z_mean = spmm(adj, hm); z_log_std = spmm(adj, hl)
  zero4_kernel<<<(2 * NN * H2D / 4 + 255) / 256, 256, 0, stream>>>((float4*)z_mean, 2 * NN * H2D / 4);
  spmm_kernel<H2D><<<EE * 32 / 256, 256, 0, stream>>>(vals, src, dst, hm, z_mean);
  spmm_kernel<H2D><<<EE * 32 / 256, 256, 0, stream>>>(vals, src, dst, hl, z_lstd);

  // 5) z = z_mean + eps * exp(z_log_std)
  reparam_kernel<<<(NN * H2D) / 256, 256, 0, stream>>>(z_mean, z_lstd, eps, z, NN * H2D);

  // 6) recon = z @ z^T
  decoder_kernel<<<dim3(NN / 16, NN / (64 * 8)), 256, 0, stream>>>(z, recon);
}